// Canny_16724602651108
// MI455X (gfx1250) — compile-verified
//
#include <hip/hip_runtime.h>
#include <hip/hip_bf16.h>

// Fused Canny pipeline for MI455X (gfx1250, wave32).
// One block = one 28x28 output tile of one batch image. All intermediates in LDS.
// Gaussian vertical pass runs on the matrix cores as a banded-matrix WMMA
// (v_wmma_f32_16x16x4_f32); halo loads use CDNA5 async global->LDS.

#define TS 28            // output tile
#define BLR 32           // blurred region rows/cols (TS+4)
#define MR  30           // magnitude region (TS+2)
#define HR  36           // h-pass rows / input rows (TS+8)
#define INW 36           // input cols

typedef float v2f __attribute__((ext_vector_type(2)));
typedef float v8f __attribute__((ext_vector_type(8)));

typedef __attribute__((address_space(1))) int* as1_i32p;
typedef __attribute__((address_space(3))) int* as3_i32p;

// normalized 5-tap Gaussian (sigma=1): g = [G0,G1,G2,G1,G0]
#define G0c 0.054488684549644946f
#define G1c 0.24420134200323332f
#define G2c 0.40261994689424635f
#define T1c 0.41421356237309503f
#define T3c 2.4142135623730951f

#if __has_builtin(__builtin_amdgcn_wmma_f32_16x16x4_f32)
#define USE_WMMA 1
#else
#define USE_WMMA 0
#endif

#if __has_builtin(__builtin_amdgcn_global_load_async_to_lds_b32) && \
    __has_builtin(__builtin_amdgcn_s_wait_asynccnt)
#define USE_ASYNC 1
#else
#define USE_ASYNC 0
#endif

__device__ __forceinline__ float gband(int d) {
  // banded Gaussian weight: g[d] for d in [0,4], else 0
  float w = 0.0f;
  w = (d == 0 || d == 4) ? G0c : w;
  w = (d == 1 || d == 3) ? G1c : w;
  w = (d == 2) ? G2c : w;
  return w;
}

__global__ __launch_bounds__(256) void canny_fused_kernel(
    const float* __restrict__ x, const int* __restrict__ lowt,
    float* __restrict__ out)
{
  __shared__ float s_in[3][HR][INW + 1];   // input halo, zero-padded
  __shared__ float s_bh[3][HR][BLR + 1];   // after horizontal Gaussian
  __shared__ float s_bl[3][BLR][BLR + 1];  // after vertical Gaussian (blurred)
  __shared__ float s_mag[MR][MR + 1];      // selected-channel magnitude
  __shared__ float s_sgx[MR][MR + 1];      // selected gx
  __shared__ float s_sgy[MR][MR + 1];      // selected gy

  const int tid = threadIdx.x;
  const int b   = blockIdx.z;
  const int Y0  = blockIdx.y * TS;
  const int X0  = blockIdx.x * TS;

  // ---------------- Stage 0: load 3 x 36 x 36 input halo (zero pad) --------
  const int NIN = 3 * HR * INW;
#if USE_ASYNC
  for (int p = tid; p < NIN; p += 256) {
    const int c  = p / (HR * INW);
    const int rm = p % (HR * INW);
    const int r  = rm / INW, cc = rm % INW;
    const int gy = Y0 - 4 + r, gx = X0 - 4 + cc;
    if ((unsigned)gy < 512u && (unsigned)gx < 512u) {
      const float* gp = x + (((size_t)b * 3 + c) * 512 + gy) * 512 + gx;
      __builtin_amdgcn_global_load_async_to_lds_b32(
          (as1_i32p)gp, (as3_i32p)&s_in[c][r][cc], 0, 0);
    }
  }
  __builtin_amdgcn_s_wait_asynccnt(0);
  __syncthreads();
  for (int p = tid; p < NIN; p += 256) {
    const int c  = p / (HR * INW);
    const int rm = p % (HR * INW);
    const int r  = rm / INW, cc = rm % INW;
    const int gy = Y0 - 4 + r, gx = X0 - 4 + cc;
    if (!((unsigned)gy < 512u && (unsigned)gx < 512u)) s_in[c][r][cc] = 0.0f;
  }
#else
  for (int p = tid; p < NIN; p += 256) {
    const int c  = p / (HR * INW);
    const int rm = p % (HR * INW);
    const int r  = rm / INW, cc = rm % INW;
    const int gy = Y0 - 4 + r, gx = X0 - 4 + cc;
    float v = 0.0f;
    if ((unsigned)gy < 512u && (unsigned)gx < 512u)
      v = x[(((size_t)b * 3 + c) * 512 + gy) * 512 + gx];
    s_in[c][r][cc] = v;
  }
#endif
  __syncthreads();

  // ---------------- Stage 1: horizontal Gaussian (VALU) --------------------
  // s_bh[c][h][n] corresponds to global (Y0-4+h, X0-2+n)
  const int NBH = 3 * HR * BLR;
  for (int p = tid; p < NBH; p += 256) {
    const int c  = p / (HR * BLR);
    const int rm = p % (HR * BLR);
    const int h  = rm / BLR, n = rm % BLR;
    const float* row = &s_in[c][h][0];
    s_bh[c][h][n] = G0c * (row[n] + row[n + 4]) +
                    G1c * (row[n + 1] + row[n + 3]) +
                    G2c * row[n + 2];
  }
  __syncthreads();

  // ---------------- Stage 2: vertical Gaussian via WMMA --------------------
  // blurred[b_][n] = sum_t g[t] * Bh[b_+t][n]  ==  Wv(16x20) x Bh(20x16),
  // Wv[m][j] = g[j-m]. 12 jobs = 3 channels x (2x2) 16x16 tiles of 32x32.
#if USE_WMMA
  {
    const int wave = tid >> 5;
    const int lane = tid & 31;
    const int n    = lane & 15;     // N (and M for A-matrix) index
    const int hi   = lane >> 4;     // 0: K in {k0,k0+1}; 1: K in {k0+2,k0+3}
    for (int job = wave; job < 12; job += 8) {   // wave-uniform => EXEC all 1s
      const int c  = job >> 2;
      const int ti = (job >> 1) & 1;
      const int tj = job & 1;
      v8f acc = {};
      for (int k0 = 0; k0 < 20; k0 += 4) {
        const int j0 = k0 + 2 * hi;
        v2f a, bm;
        a.x = gband(j0 - n);        // A[m=n][j0]
        a.y = gband(j0 + 1 - n);    // A[m=n][j0+1]
        bm.x = s_bh[c][16 * ti + j0][16 * tj + n];
        bm.y = s_bh[c][16 * ti + j0 + 1][16 * tj + n];
        acc = __builtin_amdgcn_wmma_f32_16x16x4_f32(
            false, a, false, bm, (short)0, acc, false, false);
      }
#pragma unroll
      for (int r = 0; r < 8; ++r) {
        const int M = r + 8 * hi;   // C/D layout: VGPR r -> M=r (lanes<16) / r+8
        s_bl[c][16 * ti + M][16 * tj + n] = acc[r];
      }
    }
  }
#else
  for (int p = tid; p < 3 * BLR * BLR; p += 256) {
    const int c  = p / (BLR * BLR);
    const int rm = p % (BLR * BLR);
    const int r  = rm / BLR, n = rm % BLR;
    s_bl[c][r][n] = G0c * (s_bh[c][r][n] + s_bh[c][r + 4][n]) +
                    G1c * (s_bh[c][r + 1][n] + s_bh[c][r + 3][n]) +
                    G2c * s_bh[c][r + 2][n];
  }
#endif
  __syncthreads();

  // ---------------- Stage 3: Sobel + channel argmax ------------------------
  // s_bl[c][r][n] holds blurred at global (Y0-2+r, X0-2+n).
  // Sobel reads are edge-clamped to [0,511] (reference edge-pads blurred).
  for (int p = tid; p < MR * MR; p += 256) {
    const int my = p / MR, mx = p % MR;
    const int gy = Y0 - 1 + my, gx = X0 - 1 + mx;
    float best = 0.0f, bgx = 0.0f, bgy = 0.0f;
    if ((unsigned)gy < 512u && (unsigned)gx < 512u) {
      const int r0 = min(max(gy - 1, 0), 511) - (Y0 - 2);
      const int r1 = gy - (Y0 - 2);
      const int r2 = min(max(gy + 1, 0), 511) - (Y0 - 2);
      const int c0 = min(max(gx - 1, 0), 511) - (X0 - 2);
      const int c1 = gx - (X0 - 2);
      const int c2 = min(max(gx + 1, 0), 511) - (X0 - 2);
      best = -1.0f;
#pragma unroll
      for (int c = 0; c < 3; ++c) {
        const float p00 = s_bl[c][r0][c0], p01 = s_bl[c][r0][c1], p02 = s_bl[c][r0][c2];
        const float p10 = s_bl[c][r1][c0],                        p12 = s_bl[c][r1][c2];
        const float p20 = s_bl[c][r2][c0], p21 = s_bl[c][r2][c1], p22 = s_bl[c][r2][c2];
        const float gxs = ((p02 - p00) + 2.0f * (p12 - p10) + (p22 - p20)) * 0.125f;
        const float gys = ((p20 - p00) + 2.0f * (p21 - p01) + (p22 - p02)) * 0.125f;
        const float m   = sqrtf(gxs * gxs + gys * gys + 1e-9f);
        if (m > best) { best = m; bgx = gxs; bgy = gys; }  // first-wins argmax
      }
    }
    s_mag[my][mx] = best < 0.0f ? 0.0f : best;
    s_sgx[my][mx] = bgx;
    s_sgy[my][mx] = bgy;
  }
  __syncthreads();

  // ---------------- Stage 4: NMS + threshold + sigmoid ---------------------
  const float t = (float)lowt[0];
  for (int p = tid; p < TS * TS; p += 256) {
    const int oy = p / TS, ox = p % TS;
    const int gy = Y0 + oy, gx = X0 + ox;
    if (gy < 512 && gx < 512) {
      const int my = oy + 1, mx = ox + 1;
      const float m   = s_mag[my][mx];
      const float gxv = s_sgx[my][mx];
      const float gyv = s_sgy[my][mx];
      const float gys = (gyv == 0.0f) ? 1e-9f : gyv;
      const float a   = fminf(fmaxf(gxv / gys, -10.0f), 10.0f);
      float n1, n2;
      if (a >= -T1c && a < T1c) {            // d0: "hor" masks -> up/down
        n1 = s_mag[my - 1][mx];     n2 = s_mag[my + 1][mx];
      } else if (a >= T1c && a < T3c) {      // d1: "dg1" -> up-left/down-right
        n1 = s_mag[my - 1][mx - 1]; n2 = s_mag[my + 1][mx + 1];
      } else if (a >= T3c || a < -T3c) {     // d2: "ver" -> left/right
        n1 = s_mag[my][mx - 1];     n2 = s_mag[my][mx + 1];
      } else {                               // d3: "dg2" -> up-right/down-left
        n1 = s_mag[my - 1][mx + 1]; n2 = s_mag[my + 1][mx - 1];
      }
      const float supp = (m >= n1 && m >= n2) ? m : 0.0f;
      const float v    = (supp >= t) ? supp : 1e-9f;
      out[((size_t)b * 512 + gy) * 512 + gx] = 1.0f / (1.0f + expf(-v));
    }
  }
}

extern "C" void kernel_launch(void* const* d_in, const int* in_sizes, int n_in,
                              void* d_out, int out_size, void* d_ws, size_t ws_size,
                              hipStream_t stream) {
  const float* x   = (const float*)d_in[0];
  const int* lowt  = (const int*)d_in[1];
  float* out       = (float*)d_out;
  (void)in_sizes; (void)n_in; (void)out_size; (void)d_ws; (void)ws_size;

  const int tiles = (512 + TS - 1) / TS;   // 19
  dim3 grid(tiles, tiles, 16);
  dim3 block(256, 1, 1);
  canny_fused_kernel<<<grid, block, 0, stream>>>(x, lowt, out);
}